// L2LossDif_38414187495818
// MI455X (gfx1250) — compile-verified
//
#include <hip/hip_runtime.h>

// ---------------------------------------------------------------------------
// L2LossDif for MI455X (gfx1250).
// Memory-bound streaming reduction (134 MB @ 23.3 TB/s ~ 5.8us floor) +
// a WMMA_F32_16X16X4_F32 Gram-matrix finalize on the 2048-length mu vectors.
// ---------------------------------------------------------------------------

#define D_COLS 2048
#define CG     (D_COLS / 4)      // 512 float4 column groups
#define NBLK   256               // stage-1 blocks per input
#define FWAVES 8                 // waves in finalize block
#define KCHUNK (D_COLS / FWAVES) // K per wave in finalize (256)

// workspace layout (floats)
#define MU_N_OFF   0
#define MU_A_OFF   D_COLS
#define SQ_N_OFF   (2 * D_COLS)
#define SQ_A_OFF   (2 * D_COLS + 1)
#define PART_OFF   4352                         // per-block column partials (reused per input)
#define SQPART_OFF (PART_OFF + NBLK * D_COLS)   // per-block sum-of-squares partials
#define WS_FLOATS  (SQPART_OFF + NBLK)
#define WS_FLOATS_ATOMIC (2 * D_COLS + 2)

typedef __attribute__((ext_vector_type(2))) float v2f;
typedef __attribute__((ext_vector_type(8))) float v8f;

// ---------------------------------------------------------------------------
// Stage 1 (deterministic path): stream rows, per-thread column-group sums.
// Each thread owns columns [4t, 4t+3]; block b owns a contiguous row slab.
// Emits global_load_b128 + global_prefetch_b8 (gfx1250) in the hot loop.
// ---------------------------------------------------------------------------
__global__ __launch_bounds__(512)
void stage1_colsum(const float* __restrict__ X, int rows,
                   float* __restrict__ part, float* __restrict__ sqpart) {
  const int t = threadIdx.x;                       // 0..511
  const int b = blockIdx.x;
  const int rpb = (rows + (int)gridDim.x - 1) / (int)gridDim.x;
  const int r0 = b * rpb;
  const int r1 = min(r0 + rpb, rows);
  const float4* __restrict__ Xv = (const float4*)X;

  float4 acc = make_float4(0.f, 0.f, 0.f, 0.f);
  float  sacc = 0.f;

#pragma unroll 4
  for (int r = r0; r < r1; ++r) {
    if (r + 8 < r1)
      __builtin_prefetch((const void*)&Xv[(size_t)(r + 8) * CG + t], 0, 3);
    float4 x = Xv[(size_t)r * CG + t];
    acc.x += x.x; acc.y += x.y; acc.z += x.z; acc.w += x.w;
    sacc = fmaf(x.x, x.x, sacc);
    sacc = fmaf(x.y, x.y, sacc);
    sacc = fmaf(x.z, x.z, sacc);
    sacc = fmaf(x.w, x.w, sacc);
  }

  float* p = part + (size_t)b * D_COLS + 4 * t;
  p[0] = acc.x; p[1] = acc.y; p[2] = acc.z; p[3] = acc.w;

  // fixed-order tree reduction of sum-of-squares (bitwise deterministic)
  __shared__ float red[512];
  red[t] = sacc;
  __syncthreads();
  for (int s = 256; s > 0; s >>= 1) {
    if (t < s) red[t] += red[t + s];
    __syncthreads();
  }
  if (t == 0) sqpart[b] = red[0];
}

// ---------------------------------------------------------------------------
// Stage 1 (fallback, tiny workspace): FP atomics into mu/sq directly.
// ---------------------------------------------------------------------------
__global__ __launch_bounds__(512)
void stage1_atomic(const float* __restrict__ X, int rows,
                   float* __restrict__ mu, float* __restrict__ sq) {
  const int t = threadIdx.x;
  const int b = blockIdx.x;
  const int rpb = (rows + (int)gridDim.x - 1) / (int)gridDim.x;
  const int r0 = b * rpb;
  const int r1 = min(r0 + rpb, rows);
  const float4* __restrict__ Xv = (const float4*)X;

  float4 acc = make_float4(0.f, 0.f, 0.f, 0.f);
  float  sacc = 0.f;
#pragma unroll 4
  for (int r = r0; r < r1; ++r) {
    float4 x = Xv[(size_t)r * CG + t];
    acc.x += x.x; acc.y += x.y; acc.z += x.z; acc.w += x.w;
    sacc = fmaf(x.x, x.x, sacc);
    sacc = fmaf(x.y, x.y, sacc);
    sacc = fmaf(x.z, x.z, sacc);
    sacc = fmaf(x.w, x.w, sacc);
  }
  __hip_atomic_fetch_add(&mu[4 * t + 0], acc.x, __ATOMIC_RELAXED, __HIP_MEMORY_SCOPE_AGENT);
  __hip_atomic_fetch_add(&mu[4 * t + 1], acc.y, __ATOMIC_RELAXED, __HIP_MEMORY_SCOPE_AGENT);
  __hip_atomic_fetch_add(&mu[4 * t + 2], acc.z, __ATOMIC_RELAXED, __HIP_MEMORY_SCOPE_AGENT);
  __hip_atomic_fetch_add(&mu[4 * t + 3], acc.w, __ATOMIC_RELAXED, __HIP_MEMORY_SCOPE_AGENT);

  __shared__ float red[512];
  red[t] = sacc;
  __syncthreads();
  for (int s = 256; s > 0; s >>= 1) {
    if (t < s) red[t] += red[t + s];
    __syncthreads();
  }
  if (t == 0)
    __hip_atomic_fetch_add(sq, red[0], __ATOMIC_RELAXED, __HIP_MEMORY_SCOPE_AGENT);
}

__global__ void zero_ws(float* ws, int n) {
  int i = blockIdx.x * blockDim.x + threadIdx.x;
  if (i < n) ws[i] = 0.f;
}

// ---------------------------------------------------------------------------
// Stage 2: fold NBLK block partials into mu[2048] and scalar sq (fixed order).
// ---------------------------------------------------------------------------
__global__ void stage2_reduce(const float* __restrict__ part,
                              const float* __restrict__ sqpart,
                              float* __restrict__ mu, float* __restrict__ sq) {
  int c = blockIdx.x * blockDim.x + threadIdx.x;
  if (c < D_COLS) {
    float s = 0.f;
    for (int b = 0; b < NBLK; ++b) s += part[(size_t)b * D_COLS + c];
    mu[c] = s;
  }
  if (c == 0) {
    float s = 0.f;
    for (int b = 0; b < NBLK; ++b) s += sqpart[b];
    *sq = s;
  }
}

// ---------------------------------------------------------------------------
// Finalize: 8 waves. Gram matrix of M = [mu_n; mu_a] (2 x 2048) via chained
// V_WMMA_F32_16X16X4_F32:  D += A(16x4) * B(4x16), K marching over a per-wave
// 256-length chunk.
//   A lane layout: lane = M (0..15), VGPR0/1 = K {kb, kb+1}, kb = (lane>=16)?2:0
//   B lane layout: lane = N, same K striping -> per-lane A and B are identical
//   D lane layout: VGPR v holds M=v (lanes 0-15) / M=v+8 (lanes 16-31), N=lane&15
// So D[vgpr0,lane0]=mu_n.mu_n, D[vgpr0,lane1]=mu_n.mu_a, D[vgpr1,lane1]=mu_a.mu_a.
//
// mu vectors are staged through LDS (16 KB of the 320 KB WGP LDS). Lane
// selection uses ARITHMETIC masking (av = sel_n*n + sel_a*a with per-lane 0/1
// constants) so both LDS loads are unconditionally consumed -> the compiler
// cannot predicate them; EXEC stays all-ones (WMMA requirement) and the inner
// loop is ds_load_b64 x2 + fma x4 + wmma, fully pipelined.
// ---------------------------------------------------------------------------
__global__ __launch_bounds__(32 * FWAVES)
void finalize(const float* __restrict__ ws, float* __restrict__ out,
              int nnum, int anum) {
  __shared__ float smun[D_COLS];
  __shared__ float smua[D_COLS];
  __shared__ float pnn[FWAVES], pna[FWAVES], paa[FWAVES];

  const int tid = threadIdx.x;                  // 0..255

  // Cooperative coalesced staging of mu_n / mu_a into LDS (float4 per thread).
  {
    const float4* __restrict__ mun4 = (const float4*)(ws + MU_N_OFF);
    const float4* __restrict__ mua4 = (const float4*)(ws + MU_A_OFF);
    float4* smun4 = (float4*)smun;
    float4* smua4 = (float4*)smua;
    for (int i = tid; i < D_COLS / 4; i += 32 * FWAVES) {
      smun4[i] = mun4[i];
      smua4[i] = mua4[i];
    }
  }
  __syncthreads();

  const int lane = tid & 31;
  const int wave = tid >> 5;                    // 0..7
  const int m    = lane & 15;                   // A row / B col this lane feeds
  const int kb   = (lane >> 4) << 1;            // K base within step: 0 or 2
  const int k0   = wave * KCHUNK;
  const int k1   = k0 + KCHUNK;

  // Per-lane 0/1 selection constants (computed once; no control flow in loop).
  const float sel_n = (m == 0) ? 1.f : 0.f;
  const float sel_a = (m == 1) ? 1.f : 0.f;

  v8f d = {};
  for (int k = k0; k < k1; k += 4) {
    // Unconditional 8B-aligned LDS loads (k%4==0, kb in {0,2} -> even index).
    const v2f nv = *(const v2f*)&smun[k + kb];  // ds_load_b64
    const v2f av_ = *(const v2f*)&smua[k + kb]; // ds_load_b64
    v2f a;
    a[0] = fmaf(sel_n, nv[0], sel_a * av_[0]);  // pure VALU select
    a[1] = fmaf(sel_n, nv[1], sel_a * av_[1]);
    d = __builtin_amdgcn_wmma_f32_16x16x4_f32(
        /*neg_a=*/false, a, /*neg_b=*/false, a,
        /*c_mod=*/(short)0, d, /*reuse_a=*/false, /*reuse_b=*/false);
  }

  // Per-wave partial Gram scalars.
  if (lane == 0) pnn[wave] = d[0];              // M=0,N=0
  if (lane == 1) { pna[wave] = d[0]; paa[wave] = d[1]; }  // M=0,N=1 / M=1,N=1
  __syncthreads();

  if (tid == 0) {
    double nn = 0.0, na = 0.0, aa = 0.0;
    for (int w = 0; w < FWAVES; ++w) {          // fixed order -> deterministic
      nn += (double)pnn[w];
      na += (double)pna[w];
      aa += (double)paa[w];
    }
    const double sqn = (double)ws[SQ_N_OFF];
    const double sqa = (double)ws[SQ_A_OFF];
    const double N = (double)nnum, A = (double)anum;

    const double nsum   = N * sqn - nn;
    const double asum   = A * sqa - aa;
    const double cross  = A * sqn + N * sqa - 2.0 * na;
    const double count  = N * A;
    const double ncount = N * (N - 1.0) * 0.5;
    const double acount = A * (A - 1.0) * 0.5;

    const double loss_dif = cross / count;
    const double within   = (asum + nsum) / (acount + ncount);
    const double loss     = -log(loss_dif / (loss_dif + within));
    out[0] = (float)loss;
  }
}

// ---------------------------------------------------------------------------
extern "C" void kernel_launch(void* const* d_in, const int* in_sizes, int n_in,
                              void* d_out, int out_size, void* d_ws, size_t ws_size,
                              hipStream_t stream) {
  const float* nfeats = (const float*)d_in[0];
  const float* afeats = (const float*)d_in[1];
  const int nnum = in_sizes[0] / D_COLS;
  const int anum = in_sizes[1] / D_COLS;

  float* ws   = (float*)d_ws;
  float* mun  = ws + MU_N_OFF;
  float* mua  = ws + MU_A_OFF;
  float* sqn  = ws + SQ_N_OFF;
  float* sqa  = ws + SQ_A_OFF;

  if (ws_size >= WS_FLOATS * sizeof(float)) {
    // Deterministic two-stage reduction (no FP atomics).
    float* part   = ws + PART_OFF;
    float* sqpart = ws + SQPART_OFF;

    stage1_colsum<<<NBLK, 512, 0, stream>>>(nfeats, nnum, part, sqpart);
    stage2_reduce<<<(D_COLS + 255) / 256, 256, 0, stream>>>(part, sqpart, mun, sqn);

    stage1_colsum<<<NBLK, 512, 0, stream>>>(afeats, anum, part, sqpart);
    stage2_reduce<<<(D_COLS + 255) / 256, 256, 0, stream>>>(part, sqpart, mua, sqa);
  } else {
    // Fallback: FP atomics, only ~16 KB of workspace needed.
    zero_ws<<<(WS_FLOATS_ATOMIC + 255) / 256, 256, 0, stream>>>(ws, WS_FLOATS_ATOMIC);
    stage1_atomic<<<NBLK, 512, 0, stream>>>(nfeats, nnum, mun, sqn);
    stage1_atomic<<<NBLK, 512, 0, stream>>>(afeats, anum, mua, sqa);
  }

  finalize<<<1, 32 * FWAVES, 0, stream>>>(ws, (float*)d_out, nnum, anum);
}